// ATTGCN_86517821210877
// MI455X (gfx1250) — compile-verified
//
#include <hip/hip_runtime.h>
#include <math.h>

#define Bn 64
#define Tn 128
#define Vn 25
#define Cn 128
#define Kn 128
#define Rn 32

typedef __attribute__((ext_vector_type(2))) float v2f;
typedef __attribute__((ext_vector_type(8))) float v8f;

__device__ inline v8f wmma_f32(v2f a, v2f b, v8f c) {
  // 8 args: (neg_a, A, neg_b, B, c_mod, C, reuse_a, reuse_b)
  return __builtin_amdgcn_wmma_f32_16x16x4_f32(false, a, false, b, (short)0, c,
                                               false, false);
}

// ---------- 1) xbar[b,v,c] = mean_t x[b,t,v,c] ----------
__global__ void k_mean_t(const float* __restrict__ x, float* __restrict__ xbar) {
  int idx = blockIdx.x * blockDim.x + threadIdx.x;  // B*V*C = 204800
  int c = idx & (Cn - 1);
  int v = (idx >> 7) % Vn;
  int b = (idx >> 7) / Vn;
  const float* p = x + ((size_t)(b * Tn) * Vn + v) * Cn + c;
  float s = 0.f;
  for (int t = 0; t < Tn; ++t) s += p[(size_t)t * Vn * Cn];
  xbar[idx] = s * (1.0f / Tn);
}

// ---------- generic GEMM: Out[M,128] = A[M,128] @ W[128,128] + bias ----------
// grid.x = M/16 blocks of 256 threads; wave w handles N-tile w.
__global__ void __launch_bounds__(256) k_gemm(const float* __restrict__ A,
                                              const float* __restrict__ W,
                                              const float* __restrict__ bias,
                                              float* __restrict__ Out) {
  const int tid = threadIdx.x;
  const int lane = tid & 31, wave = tid >> 5;
  const int l15 = lane & 15, half = lane >> 4;
  const int mt = blockIdx.x;
  const int nt = wave;
  const float* Arow = A + (size_t)(mt * 16 + l15) * Cn;
  v8f acc = {};
  for (int k0 = 0; k0 < Cn; k0 += 4) {
    int ka = k0 + (half << 1);
    v2f af;
    af.x = Arow[ka];
    af.y = Arow[ka + 1];
    v2f bf;
    bf.x = W[ka * Kn + nt * 16 + l15];
    bf.y = W[(ka + 1) * Kn + nt * 16 + l15];
    acc = wmma_f32(af, bf, acc);
  }
  float bv = bias[nt * 16 + l15];
  float* Ob = Out + (size_t)(mt * 16) * Kn + nt * 16 + l15;
#pragma unroll
  for (int g = 0; g < 8; ++g) {
    int m = g + (half << 3);
    Ob[(size_t)m * Kn] = acc[g] + bv;
  }
}

// ---------- F[(b,u,v),j] = tanh(phi[b,u,j] - theta[b,v,j]) ----------
__global__ void k_ftanh(const float* __restrict__ phiX,
                        const float* __restrict__ thetaX,
                        float* __restrict__ F) {
  int idx = blockIdx.x * blockDim.x + threadIdx.x;  // 40000*128
  int j = idx & (Kn - 1);
  int r = idx >> 7;
  int v = r % Vn;
  int u = (r / Vn) % Vn;
  int b = r / (Vn * Vn);
  F[idx] = tanhf(phiX[(b * Vn + u) * Kn + j] - thetaX[(b * Vn + v) * Kn + j]);
}

// ---------- At[(b,u,v),k] = A_in[k,u,v] + lam * Ahat (in place) ----------
__global__ void k_fix(float* __restrict__ At, const float* __restrict__ A_in,
                      const float* __restrict__ lam_p) {
  int idx = blockIdx.x * blockDim.x + threadIdx.x;  // B*625*128
  int k = idx & (Kn - 1);
  int r = idx >> 7;
  int v = r % Vn;
  int u = (r / Vn) % Vn;
  float lam = lam_p[0];
  At[idx] = A_in[(k * Vn + u) * Vn + v] + lam * At[idx];
}

// ---------- Zbar[b,u,k] = sum_v At[b,u,v,k] * Hbar[b,v,k] ----------
__global__ void k_zbar(const float* __restrict__ At,
                       const float* __restrict__ Hbar,
                       float* __restrict__ Zbar) {
  int idx = blockIdx.x * blockDim.x + threadIdx.x;  // B*V*K = 204800
  int k = idx & (Kn - 1);
  int u = (idx >> 7) % Vn;
  int b = (idx >> 7) / Vn;
  const float* Ar = At + ((size_t)(b * Vn + u) * Vn) * Kn + k;
  const float* Hr = Hbar + (size_t)(b * Vn) * Kn + k;
  float s = 0.f;
#pragma unroll
  for (int v = 0; v < Vn; ++v) s += Ar[v * Kn] * Hr[v * Kn];
  Zbar[idx] = s;
}

// ---------- ca[b,k] = (1/V) sum_u sum_v softmax_v(q*xk/sqrt(T)) * Hbar ----------
__global__ void k_attn_ca(const float* __restrict__ XQ,
                          const float* __restrict__ XK,
                          const float* __restrict__ Hbar,
                          float* __restrict__ ca) {
  int b = blockIdx.x;
  int k = threadIdx.x;  // 128 threads
  float xk[Vn], hb[Vn];
#pragma unroll
  for (int v = 0; v < Vn; ++v) {
    xk[v] = XK[(b * Vn + v) * Kn + k];
    hb[v] = Hbar[(b * Vn + v) * Kn + k];
  }
  const float inv_scale = 0.08838834764831845f;  // 1/sqrt(128)
  float acc = 0.f;
  for (int u = 0; u < Vn; ++u) {
    float q = XQ[(b * Vn + u) * Kn + k] * inv_scale;
    float m = -3.402823466e38f;
#pragma unroll
    for (int v = 0; v < Vn; ++v) m = fmaxf(m, q * xk[v]);
    float se = 0.f, dot = 0.f;
#pragma unroll
    for (int v = 0; v < Vn; ++v) {
      float e = expf(q * xk[v] - m);
      se += e;
      dot += e * hb[v];
    }
    acc += dot / se;
  }
  ca[b * Kn + k] = acc * (1.0f / Vn);
}

// ---------- gate[b,k] = sigmoid(gelu(LN(ca@ca1+b1)) @ ca2 + b2) ----------
__global__ void k_gate(const float* __restrict__ ca,
                       const float* __restrict__ ca1_w,
                       const float* __restrict__ ca1_b,
                       const float* __restrict__ ln1_w,
                       const float* __restrict__ ln1_b,
                       const float* __restrict__ ca2_w,
                       const float* __restrict__ ca2_b,
                       float* __restrict__ gate) {
  __shared__ float caL[Kn];
  __shared__ float y[Rn];
  __shared__ float c1[Rn];
  int b = blockIdx.x;
  int t = threadIdx.x;  // 128
  caL[t] = ca[b * Kn + t];
  __syncthreads();
  if (t < Rn) {
    float s = ca1_b[t];
    for (int kk = 0; kk < Kn; ++kk) s += caL[kk] * ca1_w[kk * Rn + t];
    y[t] = s;
  }
  __syncthreads();
  if (t < Rn) {
    float mu = 0.f;
    for (int r = 0; r < Rn; ++r) mu += y[r];
    mu *= (1.0f / Rn);
    float var = 0.f;
    for (int r = 0; r < Rn; ++r) {
      float d = y[r] - mu;
      var += d * d;
    }
    var *= (1.0f / Rn);
    float xn = (y[t] - mu) * rsqrtf(var + 1e-5f) * ln1_w[t] + ln1_b[t];
    c1[t] = 0.5f * xn * (1.0f + erff(xn * 0.7071067811865475f));  // exact gelu
  }
  __syncthreads();
  float g = ca2_b[t];
#pragma unroll
  for (int r = 0; r < Rn; ++r) g += c1[r] * ca2_w[r * Kn + t];
  gate[b * Kn + t] = 1.0f / (1.0f + expf(-g));
}

// ---------- fused: H tile via WMMA, graph-conv, gate, LayerNorm ----------
// one block per (b,t); H never touches global memory.
__global__ void __launch_bounds__(256) k_zg_ln(
    const float* __restrict__ x, const float* __restrict__ xi_w,
    const float* __restrict__ xi_b, const float* __restrict__ At,
    const float* __restrict__ gate, const float* __restrict__ ln2_w,
    const float* __restrict__ ln2_b, float* __restrict__ out) {
  __shared__ float xs[32][132];   // padded pitch: A-frag loads hit distinct banks
  __shared__ float hs[32][132];
  __shared__ float red[256];
  const int bt = blockIdx.x;
  const int b = bt / Tn, t = bt % Tn;
  const int tid = threadIdx.x;
  const int lane = tid & 31, wave = tid >> 5;
  const int l15 = lane & 15, half = lane >> 4;

  // stage x[b,t] (25 rows + 7 zero rows)
  const float* xrow = x + ((size_t)(b * Tn) + t) * Vn * Cn;
  for (int i = tid; i < 32 * Cn; i += 256) {
    int v = i >> 7, c = i & 127;
    xs[v][c] = (v < Vn) ? xrow[v * Cn + c] : 0.f;
  }
  __syncthreads();

  // H[32,128] = xs @ xi_w + xi_b ; wave -> (mt, nt0) and (mt, nt0+4)
  const int mt = wave >> 2;
  const int nt0 = wave & 3;
  v8f acc0 = {};
  v8f acc1 = {};
  for (int k0 = 0; k0 < Cn; k0 += 4) {
    int ka = k0 + (half << 1);
    v2f af;
    af.x = xs[mt * 16 + l15][ka];
    af.y = xs[mt * 16 + l15][ka + 1];
    v2f bf0, bf1;
    bf0.x = xi_w[ka * Kn + nt0 * 16 + l15];
    bf0.y = xi_w[(ka + 1) * Kn + nt0 * 16 + l15];
    bf1.x = xi_w[ka * Kn + (nt0 + 4) * 16 + l15];
    bf1.y = xi_w[(ka + 1) * Kn + (nt0 + 4) * 16 + l15];
    acc0 = wmma_f32(af, bf0, acc0);
    acc1 = wmma_f32(af, bf1, acc1);
  }
  {
    float bv0 = xi_b[nt0 * 16 + l15];
    float bv1 = xi_b[(nt0 + 4) * 16 + l15];
#pragma unroll
    for (int g = 0; g < 8; ++g) {
      int m = mt * 16 + g + (half << 3);
      hs[m][nt0 * 16 + l15] = acc0[g] + bv0;
      hs[m][(nt0 + 4) * 16 + l15] = acc1[g] + bv1;
    }
  }
  __syncthreads();

  // Z_G[u,k] = gate[k] * sum_v At[b,u,v,k]*H[v,k], then LN over k
  const int k = tid & 127;
  const int uh = tid >> 7;  // two u rows in flight
  const float gk = gate[b * Kn + k];
  const float w2 = ln2_w[k], b2 = ln2_b[k];
  for (int u0 = 0; u0 < Vn; u0 += 2) {
    int u = u0 + uh;
    float z = 0.f;
    if (u < Vn) {
      const float* Ar = At + ((size_t)(b * Vn + u) * Vn) * Kn + k;
#pragma unroll
      for (int v = 0; v < Vn; ++v) z += Ar[v * Kn] * hs[v][k];
      z *= gk;
    }
    __syncthreads();
    red[tid] = z;
    __syncthreads();
    for (int s = 64; s > 0; s >>= 1) {
      if ((tid & 127) < s) red[tid] += red[tid + s];
      __syncthreads();
    }
    float mu = red[uh << 7] * (1.0f / Kn);
    __syncthreads();
    float d = z - mu;
    red[tid] = d * d;
    __syncthreads();
    for (int s = 64; s > 0; s >>= 1) {
      if ((tid & 127) < s) red[tid] += red[tid + s];
      __syncthreads();
    }
    float var = red[uh << 7] * (1.0f / Kn);
    if (u < Vn) {
      out[(((size_t)b * Tn + t) * Vn + u) * Kn + k] =
          d * rsqrtf(var + 1e-5f) * w2 + b2;
    }
  }
}

extern "C" void kernel_launch(void* const* d_in, const int* in_sizes, int n_in,
                              void* d_out, int out_size, void* d_ws,
                              size_t ws_size, hipStream_t stream) {
  const float* x = (const float*)d_in[0];
  const float* A_in = (const float*)d_in[1];
  const float* lam = (const float*)d_in[2];
  const float* phi_w = (const float*)d_in[3];
  const float* phi_b = (const float*)d_in[4];
  const float* theta_w = (const float*)d_in[5];
  const float* theta_b = (const float*)d_in[6];
  const float* kappa_w = (const float*)d_in[7];
  const float* kappa_b = (const float*)d_in[8];
  const float* xi_w = (const float*)d_in[9];
  const float* xi_b = (const float*)d_in[10];
  const float* q_w = (const float*)d_in[11];
  const float* q_b = (const float*)d_in[12];
  const float* k_w = (const float*)d_in[13];
  const float* k_b = (const float*)d_in[14];
  const float* ca1_w = (const float*)d_in[15];
  const float* ca1_b = (const float*)d_in[16];
  const float* ln1_w = (const float*)d_in[17];
  const float* ln1_b = (const float*)d_in[18];
  const float* ca2_w = (const float*)d_in[19];
  const float* ca2_b = (const float*)d_in[20];
  const float* ln2_w = (const float*)d_in[21];
  const float* ln2_b = (const float*)d_in[22];
  float* out = (float*)d_out;

  // workspace carve (floats): total ~46.8 MB
  float* ws = (float*)d_ws;
  float* xbar = ws;                  // 204800
  float* phiX = xbar + 204800;       // 204800
  float* thetaX = phiX + 204800;     // 204800
  float* Hbar = thetaX + 204800;     // 204800
  float* Zbar = Hbar + 204800;       // 204800
  float* XQ = Zbar + 204800;         // 204800
  float* XK = XQ + 204800;           // 204800
  float* caB = XK + 204800;          // 8192
  float* gateB = caB + 8192;         // 8192
  float* F = gateB + 8192;           // 5,120,000
  float* At = F + 5120000;           // 5,120,000

  k_mean_t<<<800, 256, 0, stream>>>(x, xbar);
  k_gemm<<<100, 256, 0, stream>>>(xbar, phi_w, phi_b, phiX);     // M=1600
  k_gemm<<<100, 256, 0, stream>>>(xbar, theta_w, theta_b, thetaX);
  k_gemm<<<100, 256, 0, stream>>>(xbar, xi_w, xi_b, Hbar);
  k_ftanh<<<20000, 256, 0, stream>>>(phiX, thetaX, F);           // 40000x128
  k_gemm<<<2500, 256, 0, stream>>>(F, kappa_w, kappa_b, At);     // M=40000
  k_fix<<<20000, 256, 0, stream>>>(At, A_in, lam);
  k_zbar<<<800, 256, 0, stream>>>(At, Hbar, Zbar);
  k_gemm<<<100, 256, 0, stream>>>(Zbar, q_w, q_b, XQ);
  k_gemm<<<100, 256, 0, stream>>>(Zbar, k_w, k_b, XK);
  k_attn_ca<<<Bn, 128, 0, stream>>>(XQ, XK, Hbar, caB);
  k_gate<<<Bn, 128, 0, stream>>>(caB, ca1_w, ca1_b, ln1_w, ln1_b, ca2_w, ca2_b,
                                 gateB);
  k_zg_ln<<<Bn * Tn, 256, 0, stream>>>(x, xi_w, xi_b, At, gateB, ln2_w, ln2_b,
                                       out);
}